// ModelNew_62852551409980
// MI455X (gfx1250) — compile-verified
//
#include <hip/hip_runtime.h>

typedef float v2f __attribute__((ext_vector_type(2)));
typedef float v4f __attribute__((ext_vector_type(4)));
typedef float v8f __attribute__((ext_vector_type(8)));
typedef int   v4i __attribute__((ext_vector_type(4)));
typedef __attribute__((address_space(1))) v4i* gv4i_p;  // global int4*
typedef __attribute__((address_space(3))) v4i* lv4i_p;  // LDS int4*

#define M_DIM 16384
#define N_DIM 4096
#define K_DIM 4096
#define BM 128
#define BN 128
#define BK 16
#define LDS_STRIDE (BK + 4)   /* 20 floats = 80B: 16B aligned, conflict-free frags */
#define EP_STRIDE 132         /* 64x132 floats = 33KB epilogue buffer */
#define BUF_FLOATS ((BM + BN) * LDS_STRIDE)       /* 5120 floats per K-tile buffer */
#define SMEM_FLOATS (2 * BUF_FLOATS)              /* 10240 floats = 40KB (>= 8448) */

#if __has_builtin(__builtin_amdgcn_global_load_async_to_lds_b128)
#define ASYNC_LDS 1
#else
#define ASYNC_LDS 0
#endif

__global__ __launch_bounds__(256) void fused_gemm_gn_swish(
    const float* __restrict__ x, const float* __restrict__ w,
    const float* __restrict__ bias, const float* __restrict__ mulw,
    const float* __restrict__ gnw, const float* __restrict__ gnb,
    float* __restrict__ out)
{
    __shared__ float smem[SMEM_FLOATS];
    float* ep = smem;  // epilogue reuses buffer space after GEMM

    const int tid   = threadIdx.x;
    const int lane  = tid & 31;
    const int wave  = tid >> 5;
    const int waveM = wave & 3;            // 0..3 -> 32-row strip
    const int waveN = wave >> 2;           // 0..1 -> 64-col strip
    const int lrow  = lane & 15;
    const int half8 = (lane >> 4) << 3;    // C tile: lanes 16-31 hold rows m+8
    const int halfK = (lane >> 4) << 1;    // frag: lanes 0-15 K+{0,1}; 16-31 K+{2,3}

    const int blockN = blockIdx.x * BN;    // column-fast: W strip stays hot in L2
    const int blockM = blockIdx.y * BM;

    v8f zero = {};
    v8f acc[2][4];
    #pragma unroll
    for (int mt = 0; mt < 2; ++mt)
        #pragma unroll
        for (int nt = 0; nt < 4; ++nt)
            acc[mt][nt] = zero;

    // Stage one K-tile (A:128x16, B:128x16) into LDS buffer; 2 float4 each/thread.
    auto stage = [&](int kt, float* bufA, float* bufB) {
        #pragma unroll
        for (int i = 0; i < 2; ++i) {
            int idx = tid + i * 256;
            int r   = idx >> 2;
            int c4  = (idx & 3) << 2;
            const float* pa = x + (size_t)(blockM + r) * K_DIM + kt + c4;
            const float* pb = w + (size_t)(blockN + r) * K_DIM + kt + c4;
#if ASYNC_LDS
            __builtin_amdgcn_global_load_async_to_lds_b128(
                (gv4i_p)pa, (lv4i_p)(bufA + r * LDS_STRIDE + c4), 0, 0);
            __builtin_amdgcn_global_load_async_to_lds_b128(
                (gv4i_p)pb, (lv4i_p)(bufB + r * LDS_STRIDE + c4), 0, 0);
#else
            *(v4f*)(bufA + r * LDS_STRIDE + c4) = *(const v4f*)pa;
            *(v4f*)(bufB + r * LDS_STRIDE + c4) = *(const v4f*)pb;
#endif
            if (kt + 2 * BK < K_DIM) {
                __builtin_prefetch(pa + 2 * BK, 0, 0);  // -> global_prefetch_b8
                __builtin_prefetch(pb + 2 * BK, 0, 0);
            }
        }
    };

    auto compute = [&](const float* As, const float* Bs) {
        #pragma unroll
        for (int kk = 0; kk < BK; kk += 4) {
            v2f afr[2], bfr[4];
            #pragma unroll
            for (int mt = 0; mt < 2; ++mt)
                afr[mt] = *(const v2f*)(As + (waveM * 32 + mt * 16 + lrow) * LDS_STRIDE + kk + halfK);
            #pragma unroll
            for (int nt = 0; nt < 4; ++nt)
                bfr[nt] = *(const v2f*)(Bs + (waveN * 64 + nt * 16 + lrow) * LDS_STRIDE + kk + halfK);
            #pragma unroll
            for (int mt = 0; mt < 2; ++mt)
                #pragma unroll
                for (int nt = 0; nt < 4; ++nt)
                    acc[mt][nt] = __builtin_amdgcn_wmma_f32_16x16x4_f32(
                        false, afr[mt], false, bfr[nt],
                        (short)0, acc[mt][nt], false, false);
        }
    };

#if ASYNC_LDS
    // Double-buffered pipeline: async copies overlap current tile's WMMAs.
    stage(0, smem, smem + BM * LDS_STRIDE);
    for (int kt = 0; kt < K_DIM; kt += BK) {
        const int d = (kt / BK) & 1;
        float* As = smem + d * BUF_FLOATS;
        float* Bs = As + BM * LDS_STRIDE;
#if __has_builtin(__builtin_amdgcn_s_wait_asynccnt)
        __builtin_amdgcn_s_wait_asynccnt(0);
#else
        asm volatile("s_wait_asynccnt 0x0" ::: "memory");
#endif
        __syncthreads();   // all waves' async LDS writes for tile kt now visible
        if (kt + BK < K_DIM) {
            float* nA = smem + (d ^ 1) * BUF_FLOATS;
            stage(kt + BK, nA, nA + BM * LDS_STRIDE);
        }
        compute(As, Bs);
    }
#else
    for (int kt = 0; kt < K_DIM; kt += BK) {
        __syncthreads();
        stage(kt, smem, smem + BM * LDS_STRIDE);
        __syncthreads();
        compute(smem, smem + BM * LDS_STRIDE);
    }
#endif

    // Per-lane bias for the C-layout columns this lane owns (added before stats).
    float bv[4];
    #pragma unroll
    for (int nt = 0; nt < 4; ++nt)
        bv[nt] = bias[blockN + waveN * 64 + nt * 16 + lrow];

    // -------- Epilogue via LDS transpose: 2 passes (mt = 0,1), 64 rows each ----
    #pragma unroll
    for (int p = 0; p < 2; ++p) {
        __syncthreads();
        #pragma unroll
        for (int nt = 0; nt < 4; ++nt) {
            const int col = waveN * 64 + nt * 16 + lrow;
            #pragma unroll
            for (int r = 0; r < 8; ++r) {
                const int crow = waveM * 16 + r + half8;  // 0..63
                ep[crow * EP_STRIDE + col] = acc[p][nt][r] + bv[nt];
            }
        }
        __syncthreads();
        #pragma unroll
        for (int gi = 0; gi < 2; ++gi) {
            const int g   = tid + gi * 256;
            const int row = g >> 3;
            const int ch  = g & 7;
            float* src = ep + row * EP_STRIDE + ch * 16;
            v4f vv[4];
            #pragma unroll
            for (int q = 0; q < 4; ++q) vv[q] = *(const v4f*)(src + q * 4);
            float s = 0.f, s2 = 0.f;
            #pragma unroll
            for (int q = 0; q < 4; ++q)
                #pragma unroll
                for (int j = 0; j < 4; ++j) { float v = vv[q][j]; s += v; s2 += v * v; }
            const float mean = s * (1.0f / 16.0f);
            const float var  = s2 * (1.0f / 16.0f) - mean * mean;
            const float rstd = rsqrtf(var + 1e-6f);
            const float* gwp = gnw  + blockN + ch * 16;
            const float* gbp = gnb  + blockN + ch * 16;
            const float* mvp = mulw + blockN + ch * 16;
            #pragma unroll
            for (int q = 0; q < 4; ++q) {
                v4f gw = *(const v4f*)(gwp + q * 4);
                v4f gb = *(const v4f*)(gbp + q * 4);
                v4f mv = *(const v4f*)(mvp + q * 4);
                #pragma unroll
                for (int j = 0; j < 4; ++j) {
                    float yn = (vv[q][j] - mean) * rstd * gw[j] + gb[j];
                    float sw = yn / (1.0f + __expf(-yn));
                    float z  = sw * mv[j];
                    vv[q][j] = z / (1.0f + __expf(-z));
                }
            }
            #pragma unroll
            for (int q = 0; q < 4; ++q) *(v4f*)(src + q * 4) = vv[q];
        }
        __syncthreads();
        #pragma unroll
        for (int i = 0; i < 8; ++i) {
            const int idx = tid + i * 256;
            const int row = idx >> 5;
            const int c4  = (idx & 31) << 2;
            const int grow = blockM + (row >> 4) * 32 + p * 16 + (row & 15);
            *(v4f*)(out + (size_t)grow * N_DIM + blockN + c4) =
                *(const v4f*)(ep + row * EP_STRIDE + c4);
        }
    }
}

extern "C" void kernel_launch(void* const* d_in, const int* in_sizes, int n_in,
                              void* d_out, int out_size, void* d_ws, size_t ws_size,
                              hipStream_t stream) {
    const float* x    = (const float*)d_in[0];
    const float* w    = (const float*)d_in[1];
    const float* bias = (const float*)d_in[2];
    const float* mulw = (const float*)d_in[3];
    const float* gnw  = (const float*)d_in[4];
    const float* gnb  = (const float*)d_in[5];
    float* out = (float*)d_out;
    (void)in_sizes; (void)n_in; (void)out_size; (void)d_ws; (void)ws_size;

    dim3 grid(N_DIM / BN, M_DIM / BM); // (32, 128), column block fastest
    dim3 block(256);
    fused_gemm_gn_swish<<<grid, block, 0, stream>>>(x, w, bias, mulw, gnw, gnb, out);
}